// LSA_20392504721879
// MI455X (gfx1250) — compile-verified
//
#include <hip/hip_runtime.h>

// Problem constants (from reference setup_inputs)
#define Bb   4
#define Cch  128
#define Hh   64
#define Ww   128
#define HW   (Hh * Ww)   // 8192
#define Lwin 25          // 5x5 window

typedef __attribute__((ext_vector_type(2))) float v2f;
typedef __attribute__((ext_vector_type(8))) float v8f;

// -------------------------------------------------------------------------
// Kernel 1: v[b,o,h,w] = sum_c Wv[o,c] * fmap[b,c,h,w]
// GEMM per batch: M=128 (out ch), K=128 (in ch), N=8192 (pixels).
// One wave -> one 16x16 (MxN) tile; 8 waves per block cover all M for the
// same 16 pixels so B-fragment loads hit L0. Uses V_WMMA_F32_16X16X4_F32.
//
// A 16x4 f32 layout (ISA 7.12.2): lanes 0-15 -> M rows, VGPR0 holds K=k0
// (lanes 0-15) / K=k0+2 (lanes 16-31), VGPR1 holds k0+1 / k0+3.
// B 4x16 mirrors: lanes carry N, lane-half selects K row pair.
// D 16x16 f32: VGPR r -> row m0+r (lanes 0-15) / m0+r+8 (lanes 16-31).
// -------------------------------------------------------------------------
__global__ __launch_bounds__(256) void lsa_gemm_wmma(
    const float* __restrict__ Wv,    // [C, C]
    const float* __restrict__ fmap,  // [B, C, H, W]
    float* __restrict__ v)           // [B, C, H, W] workspace
{
    const int lane = threadIdx.x & 31;
    const int wave = threadIdx.x >> 5;
    const int b    = blockIdx.y;
    const int n0   = blockIdx.x << 4;   // pixel tile base within batch
    const int m0   = wave << 4;         // output-channel tile base

    const int Ma   = lane & 15;         // A row (M) held by this lane
    const int Nl   = lane & 15;         // B/D column (N) held by this lane
    const int koff = (lane >> 4) << 1;  // 0 for lanes 0-15, 2 for lanes 16-31

    const float* __restrict__ wrow = Wv + (m0 + Ma) * Cch;
    const float* __restrict__ bcol = fmap + (size_t)b * Cch * HW + n0 + Nl;

    v8f acc = {};
    #pragma unroll 4
    for (int k0 = 0; k0 < Cch; k0 += 4) {
        v2f a, bf;
        a.x  = wrow[k0 + koff];
        a.y  = wrow[k0 + koff + 1];
        bf.x = bcol[(size_t)(k0 + koff) * HW];
        bf.y = bcol[(size_t)(k0 + koff + 1) * HW];
        // (neg_a, A, neg_b, B, c_mod, C, reuse_a, reuse_b)
        acc = __builtin_amdgcn_wmma_f32_16x16x4_f32(
            false, a, false, bf, (short)0, acc, false, false);
    }

    const int rowsel = (lane >> 4) << 3;  // +8 rows for upper lane half
    float* __restrict__ vb = v + (size_t)b * Cch * HW;
    #pragma unroll
    for (int r = 0; r < 8; ++r) {
        vb[(size_t)(m0 + r + rowsel) * HW + n0 + Nl] = acc[r];
    }
}

// -------------------------------------------------------------------------
// Kernel 2: out[b,c,y,x] = fmap[b,c,y,x]
//                        + gamma * sum_{l=25} attn[b, y*W+x, l] * v[b,c,ny,nx]
// with (ny,nx) = replicate-clamped (y+dy-2, x+dx-2), l = dy*5+dx.
// Block = 64 x-positions (coalesced v reads & out writes) x 4 channels.
// attn tile for the 64 pixels is staged once in LDS and shared by channels.
// -------------------------------------------------------------------------
__global__ __launch_bounds__(256) void lsa_aggregate(
    const float* __restrict__ attn,   // [B, 1, H*W, 25]
    const float* __restrict__ fmap,   // [B, C, H, W]
    const float* __restrict__ v,      // [B, C, H, W]
    const float* __restrict__ gamma,  // [1]
    float* __restrict__ out)          // [B, C, H, W]
{
    __shared__ float sAttn[64 * Lwin];

    const int tx    = threadIdx.x;        // 0..63 : x within tile
    const int ty    = threadIdx.y;        // 0..3  : channel within tile
    const int xbase = blockIdx.x << 6;
    const int y     = blockIdx.y;
    const int b     = blockIdx.z >> 5;
    const int ctile = blockIdx.z & 31;

    // Stage attn[b, y*W + xbase .. +63, 0..24] into LDS (contiguous loads).
    const int tid   = ty * 64 + tx;
    const size_t abase = ((size_t)b * HW + (size_t)y * Ww + xbase) * Lwin;
    for (int idx = tid; idx < 64 * Lwin; idx += 256) {
        sAttn[idx] = attn[abase + idx];
    }
    __syncthreads();

    const int x = xbase + tx;
    const int c = (ctile << 2) + ty;

    const float* __restrict__ aw = sAttn + tx * Lwin;
    float acc = 0.0f;
    #pragma unroll
    for (int dy = 0; dy < 5; ++dy) {
        int ny = y + dy - 2;
        ny = ny < 0 ? 0 : (ny > Hh - 1 ? Hh - 1 : ny);
        const float* __restrict__ vrow =
            v + (((size_t)b * Cch + c) * Hh + ny) * Ww;
        #pragma unroll
        for (int dx = 0; dx < 5; ++dx) {
            int nx = x + dx - 2;
            nx = nx < 0 ? 0 : (nx > Ww - 1 ? Ww - 1 : nx);
            acc = fmaf(aw[dy * 5 + dx], vrow[nx], acc);
        }
    }

    const size_t o = (((size_t)b * Cch + c) * Hh + y) * Ww + x;
    out[o] = fmaf(gamma[0], acc, fmap[o]);
}

// -------------------------------------------------------------------------
// Launch
// -------------------------------------------------------------------------
extern "C" void kernel_launch(void* const* d_in, const int* in_sizes, int n_in,
                              void* d_out, int out_size, void* d_ws, size_t ws_size,
                              hipStream_t stream) {
    const float* attn  = (const float*)d_in[0];  // [4,1,8192,25]
    const float* fmap  = (const float*)d_in[1];  // [4,128,64,128]
    const float* Wv    = (const float*)d_in[2];  // [128,128]
    const float* gamma = (const float*)d_in[3];  // [1]
    float* out = (float*)d_out;                  // [4,128,64,128]
    float* v   = (float*)d_ws;                   // [4,128,64,128] = 16.8 MB

    // Kernel 1: 512 pixel-tiles per batch x 4 batches; 8 waves/block (all M).
    dim3 g1(HW / 16, Bb, 1);
    dim3 b1(256, 1, 1);
    lsa_gemm_wmma<<<g1, b1, 0, stream>>>(Wv, fmap, v);

    // Kernel 2: x-tiles (2) x rows (64) x (batch * channel-tiles = 4*32).
    dim3 g2(Ww / 64, Hh, Bb * (Cch / 4));
    dim3 b2(64, 4, 1);
    lsa_aggregate<<<g2, b2, 0, stream>>>(attn, fmap, v, gamma, out);
}